// Wav2VecLoss_30803505446946
// MI455X (gfx1250) — compile-verified
//
#include <hip/hip_runtime.h>

// Problem constants (match reference)
#define TT    4096
#define CC    512
#define NB    2
#define KSTEPS 12
#define NNEG  10

typedef __attribute__((ext_vector_type(2))) float v2f;
typedef __attribute__((ext_vector_type(8))) float v8f;

// Numerically stable log(sigmoid(x)) = min(x,0) - log(1 + exp(-|x|))
__device__ __forceinline__ float log_sigmoid(float x) {
    float m = fminf(x, 0.0f);
    return m - __logf(1.0f + __expf(-fabsf(x)));
}

__device__ __forceinline__ float wave_reduce(float v) {
    #pragma unroll
    for (int off = 16; off > 0; off >>= 1)
        v += __shfl_xor(v, off, 32);
    return v;
}

__global__ void init_out_kernel(float* out) {
    if (threadIdx.x == 0) out[0] = 0.0f;
}

// One workgroup (8 wave32) per (batch b, 16-wide s-tile of c).
// Stage 1: c[:, s0:s0+16] = W * ctx[b,:,s0:s0+16] + bias  -> LDS (WMMA f32 16x16x4)
// Stage 2: wave0/1: band tiles cont[t0..t0+15, s-tile] for the 12 diagonals (pos loss)
//          wave2:   cont[0..15, s-tile] -> negrows buffer (rows 0..9 used later)
__global__ __launch_bounds__(256) void fused_wav2vec_kernel(
    const float* __restrict__ enc,     // [B,C,T]
    const float* __restrict__ ctx,     // [B,C,T]
    const float* __restrict__ W,       // [C,C] row-major (d,c)
    const float* __restrict__ bias,    // [C]
    float* __restrict__ negrows,       // [B,16,T] scratch
    float* __restrict__ out_accum)     // scalar accumulator
{
    __shared__ float c_lds[CC][16];    // 32 KB

    const int b    = blockIdx.y;
    const int s0   = blockIdx.x * 16;
    const int lane = threadIdx.x & 31;
    const int wave = threadIdx.x >> 5;
    const int ls   = lane & 15;        // M / N index within tile
    const int hi   = lane >> 4;        // half-wave: handles K+2,K+3
    const int kOff = 2 * hi;

    const float* ctxb = ctx + (size_t)b * CC * TT;
    const float* encb = enc + (size_t)b * CC * TT;

    // ---------------- Stage 1: column block of c ----------------
    v8f acc[4] = {};
    for (int kk = 0; kk < CC; kk += 4) {
        // B operand (4x16, K x N): lanes 0-15 hold K=kk+0/+1, lanes 16-31 K=kk+2/+3
        const float* bp = ctxb + (size_t)(kk + kOff) * TT + s0 + ls;
        v2f bmat;
        bmat.x = bp[0];
        bmat.y = bp[TT];
        #pragma unroll
        for (int j = 0; j < 4; ++j) {
            const int m0 = (wave * 4 + j) * 16;
            // A operand (16x4, M x K): row m0+ls, cols kk+kOff, kk+kOff+1 (contiguous pair)
            v2f amat = *(const v2f*)(W + (size_t)(m0 + ls) * CC + kk + kOff);
            acc[j] = __builtin_amdgcn_wmma_f32_16x16x4_f32(
                false, amat, false, bmat, (short)0, acc[j], false, false);
        }
    }
    // bias add + write D tiles to LDS (D layout: VGPR r -> rows r / r+8 per half-wave)
    #pragma unroll
    for (int j = 0; j < 4; ++j) {
        const int m0 = (wave * 4 + j) * 16;
        #pragma unroll
        for (int r = 0; r < 8; ++r) {
            const int m = m0 + r + hi * 8;
            c_lds[m][ls] = acc[j][r] + bias[m];
        }
    }
    __syncthreads();

    // ---------------- Stage 2: band (pos) + neg rows ----------------
    if (wave < 3) {
        const int t0 = (wave == 2) ? 0 : (s0 + wave * 16);
        v8f d = {};
        for (int kk = 0; kk < CC; kk += 4) {
            // A operand: enc column t = t0+ls, K rows kk+kOff, kk+kOff+1
            const int t = t0 + ls;
            v2f amat;
            if (t < TT) {   // per-lane load guard only; WMMA itself stays EXEC=all-1s
                const float* ap = encb + (size_t)(kk + kOff) * TT + t;
                amat.x = ap[0];
                amat.y = ap[TT];
            } else {
                amat.x = 0.0f; amat.y = 0.0f;
            }
            // B operand: c tile from LDS
            v2f bmat;
            bmat.x = c_lds[kk + kOff][ls];
            bmat.y = c_lds[kk + kOff + 1][ls];
            d = __builtin_amdgcn_wmma_f32_16x16x4_f32(
                false, amat, false, bmat, (short)0, d, false, false);
        }

        if (wave == 2) {
            // cont[b, 0..15, s0+ls] -> negrows
            #pragma unroll
            for (int r = 0; r < 8; ++r) {
                const int rr = r + hi * 8;
                negrows[((size_t)b * 16 + rr) * TT + s0 + ls] = d[r];
            }
        } else {
            // positives: elements with 0 <= (t - s) < K_STEPS, t < T
            float lsum = 0.0f;
            #pragma unroll
            for (int r = 0; r < 8; ++r) {
                const int t  = t0 + r + hi * 8;
                const int s  = s0 + ls;
                const int dd = t - s;
                if (dd >= 0 && dd < KSTEPS && t < TT)
                    lsum += log_sigmoid(d[r]);
            }
            lsum = wave_reduce(lsum);
            if (lane == 0) atomicAdd(out_accum, lsum);
        }
    }
}

// negatives: neg[b,n,t] (step k) = cont[b, n, neg_idx[k,t,n]]; loss += NUM_NEG * logsig(-v)
__global__ __launch_bounds__(256) void neg_loss_kernel(
    const float* __restrict__ negrows,   // [B,16,T]
    const int*   __restrict__ neg_idx,   // [K,T,NNEG]
    float* __restrict__ out_accum)
{
    const int gid = blockIdx.x * blockDim.x + threadIdx.x;   // k*T + t
    float lsum = 0.0f;
    if (gid < KSTEPS * TT) {
        const int k = gid / TT;
        const int t = gid - k * TT;
        const int* ip = neg_idx + ((size_t)k * TT + t) * NNEG;
        #pragma unroll
        for (int n = 0; n < NNEG; ++n) {
            const int idx = ip[n];
            #pragma unroll
            for (int b = 0; b < NB; ++b) {
                const float v = negrows[((size_t)b * 16 + n) * TT + idx];
                lsum += log_sigmoid(-v);
            }
        }
        lsum *= (float)NNEG;
    }
    lsum = wave_reduce(lsum);
    if ((threadIdx.x & 31) == 0) atomicAdd(out_accum, lsum);
}

extern "C" void kernel_launch(void* const* d_in, const int* in_sizes, int n_in,
                              void* d_out, int out_size, void* d_ws, size_t ws_size,
                              hipStream_t stream) {
    const float* enc     = (const float*)d_in[0];   // feat_enc   [B,C,T]
    const float* ctx     = (const float*)d_in[1];   // feat_context [B,C,T]
    const float* W       = (const float*)d_in[2];   // [C,C]
    const float* bias    = (const float*)d_in[3];   // [C]
    const int*   neg_idx = (const int*)d_in[4];     // [K,T,NNEG]
    float* out     = (float*)d_out;
    float* negrows = (float*)d_ws;                  // B*16*T floats = 512 KB

    init_out_kernel<<<1, 32, 0, stream>>>(out);

    dim3 grid(TT / 16, NB);
    fused_wav2vec_kernel<<<grid, 256, 0, stream>>>(enc, ctx, W, bias, negrows, out);

    const int nthreads = KSTEPS * TT;
    neg_loss_kernel<<<(nthreads + 255) / 256, 256, 0, stream>>>(negrows, neg_idx, out);
}